// CriticGCN_36094905155711
// MI455X (gfx1250) — compile-verified
//
#include <hip/hip_runtime.h>

typedef __attribute__((ext_vector_type(2))) float v2f;
typedef __attribute__((ext_vector_type(8))) float v8f;

#define TPB 256

// ---------------------------------------------------------------------------
// Kernel 1: fold the head into the first layer.
//   block b <  F : w[b] = dot(W1[b, :], Wh)          (F dot products of len H)
//   block b == F : c    = dot(b1, Wh) + bh
// ---------------------------------------------------------------------------
__global__ __launch_bounds__(TPB)
void fold_head_kernel(const float* __restrict__ W1,   // [F, H]
                      const float* __restrict__ b1,   // [H]
                      const float* __restrict__ Wh,   // [H]
                      const float* __restrict__ bh,   // [1]
                      float* __restrict__ w,          // [F]
                      float* __restrict__ cconst,     // [1]
                      int F, int H) {
    __shared__ float red[TPB];
    const int b = blockIdx.x;
    const float* vec = (b < F) ? (W1 + (size_t)b * H) : b1;
    float acc = 0.0f;
    for (int j = threadIdx.x; j < H; j += TPB) acc += vec[j] * Wh[j];
    red[threadIdx.x] = acc;
    __syncthreads();
    for (int off = TPB / 2; off > 0; off >>= 1) {
        if (threadIdx.x < off) red[threadIdx.x] += red[threadIdx.x + off];
        __syncthreads();
    }
    if (threadIdx.x == 0) {
        if (b < F) w[b] = red[0];
        else       cconst[0] = red[0] + bh[0];
    }
}

// ---------------------------------------------------------------------------
// Kernel 2: deg[n] = 1.0 (self loop contribution)
// ---------------------------------------------------------------------------
__global__ __launch_bounds__(TPB)
void deg_init_kernel(float* __restrict__ deg, int N) {
    int n = blockIdx.x * TPB + threadIdx.x;
    if (n < N) deg[n] = 1.0f;
}

// ---------------------------------------------------------------------------
// Kernel 3: deg[dst[e]] += 1 for each edge
// ---------------------------------------------------------------------------
__global__ __launch_bounds__(TPB)
void deg_count_kernel(const int* __restrict__ dst, float* __restrict__ deg, int E) {
    int e = blockIdx.x * TPB + threadIdx.x;
    if (e < E) atomicAdd(&deg[dst[e]], 1.0f);
}

// ---------------------------------------------------------------------------
// Kernel 4: s[n] = dot(X[n, 0:F], w) via V_WMMA_F32_16X16X4_F32.
// One wave (32 lanes) per 16 rows. A tile = 16x4 fp32 rows of X per ISA
// layout (lanes 0-15: K=kc+0/kc+1, lanes 16-31: K=kc+2/kc+3). B = w[k]
// broadcast across all 16 columns, so every D column holds the row dot
// product. Accumulate over K in C. EXEC stays all-ones (loads use clamped
// row indices, never predication) as WMMA requires.
// ---------------------------------------------------------------------------
__global__ __launch_bounds__(32)
void xw_wmma_kernel(const float* __restrict__ X,   // [N, F]
                    const float* __restrict__ w,   // [F], F % 4 == 0
                    float* __restrict__ s,         // [N]
                    int N, int F) {
    const int lane  = threadIdx.x;          // 0..31
    const int base  = blockIdx.x * 16;
    const int m     = lane & 15;            // row within tile
    const int khalf = (lane >> 4) * 2;      // 0 for lanes 0-15, 2 for 16-31
    int row = base + m;
    if (row >= N) row = N - 1;              // clamp: keep EXEC all-ones
    const float* xr = X + (size_t)row * F;

    v8f c = {};
    for (int kc = 0; kc < F; kc += 4) {
        const int k0 = kc + khalf;
        v2f a, b;
        a.x = xr[k0];
        a.y = xr[k0 + 1];
        b.x = w[k0];                        // broadcast over all N columns
        b.y = w[k0 + 1];
        c = __builtin_amdgcn_wmma_f32_16x16x4_f32(
                /*neg_a=*/false, a, /*neg_b=*/false, b,
                /*c_mod=*/(short)0, c, /*reuse_a=*/false, /*reuse_b=*/false);
    }

    // D[m][n] identical for all n. Lane 0 holds M=0..7 in c[0..7],
    // lane 16 holds M=8..15.
    if ((lane & 15) == 0) {
        const int mb = (lane >> 4) * 8;
        if (lane == 0 || lane == 16) {
            #pragma unroll
            for (int r = 0; r < 8; ++r) {
                const int rr = base + mb + r;
                if (rr < N) s[rr] = c[r];
            }
        }
    }
}

// ---------------------------------------------------------------------------
// Kernel 5: dinv[n] = rsqrt(deg[n]);  acc[n] = dinv^2 * s[n]  (self loop)
// ---------------------------------------------------------------------------
__global__ __launch_bounds__(TPB)
void dinv_self_kernel(const float* __restrict__ deg,
                      const float* __restrict__ s,
                      float* __restrict__ dinv,
                      float* __restrict__ acc, int N) {
    int n = blockIdx.x * TPB + threadIdx.x;
    if (n < N) {
        float di = rsqrtf(deg[n]);   // deg >= 1 always (self loops)
        dinv[n] = di;
        acc[n]  = di * di * s[n];
    }
}

// ---------------------------------------------------------------------------
// Kernel 6: acc[dst] += dinv[src]*dinv[dst]*s[src]  (scalar edge scatter)
// ---------------------------------------------------------------------------
__global__ __launch_bounds__(TPB)
void scatter_kernel(const int* __restrict__ src,
                    const int* __restrict__ dst,
                    const float* __restrict__ dinv,
                    const float* __restrict__ s,
                    float* __restrict__ acc, int E) {
    int e = blockIdx.x * TPB + threadIdx.x;
    if (e < E) {
        const int u = src[e];
        const int v = dst[e];
        atomicAdd(&acc[v], dinv[u] * dinv[v] * s[u]);
    }
}

// ---------------------------------------------------------------------------
// Kernel 7: out[n] = acc[n] + c
// ---------------------------------------------------------------------------
__global__ __launch_bounds__(TPB)
void finalize_kernel(const float* __restrict__ acc,
                     const float* __restrict__ cconst,
                     float* __restrict__ out, int N) {
    int n = blockIdx.x * TPB + threadIdx.x;
    if (n < N) out[n] = acc[n] + cconst[0];
}

// ---------------------------------------------------------------------------
extern "C" void kernel_launch(void* const* d_in, const int* in_sizes, int n_in,
                              void* d_out, int out_size, void* d_ws, size_t ws_size,
                              hipStream_t stream) {
    const float* state = (const float*)d_in[0];   // [N, F, 1]
    const int*   eidx  = (const int*)  d_in[1];   // [2, E]
    const float* W1    = (const float*)d_in[2];   // [F, H]
    const float* b1    = (const float*)d_in[3];   // [H]
    const float* Wh    = (const float*)d_in[4];   // [H, 1]
    const float* bh    = (const float*)d_in[5];   // [1]
    float*       out   = (float*)d_out;           // [N, 1]

    const int H = in_sizes[3];                    // 1024
    const int F = in_sizes[2] / H;                // 100
    const int N = in_sizes[0] / F;                // 50000
    const int E = in_sizes[1] / 2;                // 200000
    const int* e_src = eidx;
    const int* e_dst = eidx + E;

    // workspace layout (floats)
    float* ws   = (float*)d_ws;
    float* w    = ws;                 // [F]   (reserve 256)
    float* cc   = ws + 255;           // [1]
    float* deg  = ws + 256;           // [N]
    float* dinv = deg + N;            // [N]
    float* s    = dinv + N;           // [N]
    float* acc  = s + N;              // [N]

    const int gN = (N + TPB - 1) / TPB;
    const int gE = (E + TPB - 1) / TPB;
    const int gW = (N + 15) / 16;

    fold_head_kernel<<<F + 1, TPB, 0, stream>>>(W1, b1, Wh, bh, w, cc, F, H);
    deg_init_kernel <<<gN,    TPB, 0, stream>>>(deg, N);
    deg_count_kernel<<<gE,    TPB, 0, stream>>>(e_dst, deg, E);
    xw_wmma_kernel  <<<gW,    32,  0, stream>>>(state, w, s, N, F);
    dinv_self_kernel<<<gN,    TPB, 0, stream>>>(deg, s, dinv, acc, N);
    scatter_kernel  <<<gE,    TPB, 0, stream>>>(e_src, e_dst, dinv, s, acc, E);
    finalize_kernel <<<gN,    TPB, 0, stream>>>(acc, cc, out, N);
}